// Attn_fuser_1881195676147
// MI455X (gfx1250) — compile-verified
//
#include <hip/hip_runtime.h>

// Problem constants (from reference)
#define NL 4
#define DEMB 256
#define NH 2
#define DH 128
#define NFACE 4096
#define NEDGE 8192
#define ELOOP 32

typedef __attribute__((ext_vector_type(2))) float v2f;
typedef __attribute__((ext_vector_type(8))) float v8f;

// ---------------------------------------------------------------------------
// GEMM: C[M,256] = A[M,256] * W[256,256]^T + bias   (all fp32, WMMA f32 path)
// One wave per 16x32 C tile (two 16x16 n-tiles sharing the A fragment).
// K split into even/odd sub-chains -> 4 independent WMMA accumulator chains.
// A fragment per lane: m = lane%16, k = k0 + (lane/16)*2 + {0,1}  (float2 load)
// B fragment per lane: n = lane%16, same k pair -> contiguous in W row (W is [out,in])
// C/D layout: vgpr r -> M = r + (lane/16)*8, N = lane%16
// ---------------------------------------------------------------------------
__global__ __launch_bounds__(256) void gemm_xwT_kernel(
    const float* __restrict__ A,
    const float* __restrict__ W,
    const float* __restrict__ bias,
    float* __restrict__ C,
    int M)
{
    const int lane = threadIdx.x & 31;
    const int wave = threadIdx.x >> 5;
    const int tile = blockIdx.x * 8 + wave;     // grids sized so tile is in range
    const int tn   = (tile & 7) * 32;           // 256/32 = 8 n-tiles (32 wide)
    const int tm   = (tile >> 3) * 16;
    const int half = lane >> 4;                 // 0 or 1
    const int lid  = lane & 15;
    const int kb   = half * 2;

    const float* arow  = A + (size_t)(tm + lid) * DEMB;       // lane's A row (m = lid)
    const float* wrow0 = W + (size_t)(tn + lid) * DEMB;       // n-tile 0 row
    const float* wrow1 = W + (size_t)(tn + 16 + lid) * DEMB;  // n-tile 1 row

    v8f acc00 = {}, acc01 = {}, acc10 = {}, acc11 = {};
#pragma unroll 2
    for (int k0 = 0; k0 < DEMB; k0 += 8) {
        v2f a0  = *(const v2f*)(arow  + k0     + kb);
        v2f a1  = *(const v2f*)(arow  + k0 + 4 + kb);
        v2f b00 = *(const v2f*)(wrow0 + k0     + kb);
        v2f b01 = *(const v2f*)(wrow0 + k0 + 4 + kb);
        v2f b10 = *(const v2f*)(wrow1 + k0     + kb);
        v2f b11 = *(const v2f*)(wrow1 + k0 + 4 + kb);
        acc00 = __builtin_amdgcn_wmma_f32_16x16x4_f32(false, a0, false, b00, (short)0, acc00, false, false);
        acc10 = __builtin_amdgcn_wmma_f32_16x16x4_f32(false, a0, false, b10, (short)0, acc10, false, false);
        acc01 = __builtin_amdgcn_wmma_f32_16x16x4_f32(false, a1, false, b01, (short)0, acc01, false, false);
        acc11 = __builtin_amdgcn_wmma_f32_16x16x4_f32(false, a1, false, b11, (short)0, acc11, false, false);
    }

    const float bn0 = bias[tn + lid];
    const float bn1 = bias[tn + 16 + lid];
    float* crow0 = C + (size_t)(tm + half * 8) * DEMB + tn + lid;
    float* crow1 = crow0 + 16;
#pragma unroll
    for (int r = 0; r < 8; ++r) {
        crow0[(size_t)r * DEMB] = acc00[r] + acc01[r] + bn0;
        crow1[(size_t)r * DEMB] = acc10[r] + acc11[r] + bn1;
    }
}

// ---------------------------------------------------------------------------
// Attention: one wave per (face, head). 32 candidate edges per face.
// Duplicate edge indices collapse (first occurrence wins) to match the
// reference's scatter-built mask. Masked-out faces -> all -inf -> NaN (matches
// jax softmax of an all -inf row).
// ---------------------------------------------------------------------------
__global__ __launch_bounds__(256) void attn_kernel(
    const long long* __restrict__ loop,      // [NFACE, 32] int64
    const unsigned char* __restrict__ fmask, // [NFACE] bool
    const float* __restrict__ Q,             // [NFACE, 256]
    const float* __restrict__ K,             // [NEDGE, 256]
    const float* __restrict__ V,             // [NEDGE, 256]
    float* __restrict__ AO)                  // [NFACE, 256]
{
    const int lane = threadIdx.x & 31;
    const int wave = threadIdx.x >> 5;
    const int wid  = blockIdx.x * 8 + wave;  // 0 .. NFACE*NH-1
    const int face = wid >> 1;
    const int head = wid & 1;
    const int base = head * DH + lane * 4;   // this lane's 4 dims of the head

    const int e = (int)loop[(size_t)face * ELOOP + lane];  // lane-per-edge
    const float4 qv = *(const float4*)(Q + (size_t)face * DEMB + base);

    // ---- scores: lane j ends holding score for edge j ----
    float score = 0.0f;
    for (int j = 0; j < ELOOP; ++j) {
        int ej = __shfl(e, j, 32);
        float4 kv = *(const float4*)(K + (size_t)ej * DEMB + base);
        float p = qv.x * kv.x + qv.y * kv.y + qv.z * kv.z + qv.w * kv.w;
#pragma unroll
        for (int off = 16; off > 0; off >>= 1)
            p += __shfl_xor(p, off, 32);
        if (lane == j) score = p;
    }
    score *= 0.08838834764831845f;  // 1/sqrt(128)

    // ---- dedup: drop later occurrences of the same edge ----
    bool dup = false;
    for (int i = 0; i < ELOOP; ++i) {
        int ei = __shfl(e, i, 32);
        if (i < lane && ei == e) dup = true;
    }
    if (dup || !fmask[face]) score = -__builtin_inff();

    // ---- softmax across the 32 lanes ----
    float mx = score;
#pragma unroll
    for (int off = 16; off > 0; off >>= 1)
        mx = fmaxf(mx, __shfl_xor(mx, off, 32));
    float ex = expf(score - mx);
    float sm = ex;
#pragma unroll
    for (int off = 16; off > 0; off >>= 1)
        sm += __shfl_xor(sm, off, 32);
    const float w = ex / sm;

    // ---- weighted V gather (lane-per-dim, coalesced rows) ----
    float4 acc = make_float4(0.f, 0.f, 0.f, 0.f);
    for (int j = 0; j < ELOOP; ++j) {
        float wj = __shfl(w, j, 32);
        int   ej = __shfl(e, j, 32);
        float4 vv = *(const float4*)(V + (size_t)ej * DEMB + base);
        acc.x += wj * vv.x; acc.y += wj * vv.y;
        acc.z += wj * vv.z; acc.w += wj * vv.w;
    }
    *(float4*)(AO + (size_t)face * DEMB + base) = acc;
}

// ---------------------------------------------------------------------------
// Launch: per layer -> Q,K,V GEMMs, attention, out-proj. X ping-pongs in ws.
// ws layout (floats): Q[1M] K[2M] V[2M] AO[1M] XA[1M] XB[1M]  = 32 MB total.
// ---------------------------------------------------------------------------
extern "C" void kernel_launch(void* const* d_in, const int* in_sizes, int n_in,
                              void* d_out, int out_size, void* d_ws, size_t ws_size,
                              hipStream_t stream)
{
    const long long*     loop  = (const long long*)d_in[0];      // [L,32] int64
    const unsigned char* fmask = (const unsigned char*)d_in[1];  // [L] bool
    const float*         edgeE = (const float*)d_in[2];          // [S,256]
    const float*         faceE = (const float*)d_in[3];          // [L,256]
    const float*         inW   = (const float*)d_in[4];          // [4,768,256]
    const float*         inB   = (const float*)d_in[5];          // [4,768]
    const float*         outW  = (const float*)d_in[6];          // [4,256,256]
    const float*         outB  = (const float*)d_in[7];          // [4,256]
    float*               out   = (float*)d_out;                  // [L,256]

    float* ws = (float*)d_ws;
    float* Qb = ws;
    float* Kb = Qb + (size_t)NFACE * DEMB;
    float* Vb = Kb + (size_t)NEDGE * DEMB;
    float* AO = Vb + (size_t)NEDGE * DEMB;
    float* XA = AO + (size_t)NFACE * DEMB;
    float* XB = XA + (size_t)NFACE * DEMB;
    float* xbuf[2] = { XA, XB };

    // grid sizes: tiles = (M/16)*(256/32), 8 waves (tiles) per 256-thread block
    const int gridQ = (NFACE / 16) * (DEMB / 32) / 8;  // 256
    const int gridK = (NEDGE / 16) * (DEMB / 32) / 8;  // 512
    const int gridA = NFACE * NH / 8;                  // 1024

    const float* Xcur = faceE;
    for (int l = 0; l < NL; ++l) {
        const float* Wl = inW + (size_t)l * 3 * DEMB * DEMB;
        const float* Bl = inB + (size_t)l * 3 * DEMB;

        gemm_xwT_kernel<<<gridQ, 256, 0, stream>>>(Xcur,  Wl,                    Bl,            Qb, NFACE);
        gemm_xwT_kernel<<<gridK, 256, 0, stream>>>(edgeE, Wl + DEMB * DEMB,      Bl + DEMB,     Kb, NEDGE);
        gemm_xwT_kernel<<<gridK, 256, 0, stream>>>(edgeE, Wl + 2 * DEMB * DEMB,  Bl + 2 * DEMB, Vb, NEDGE);

        attn_kernel<<<gridA, 256, 0, stream>>>(loop, fmask, Qb, Kb, Vb, AO);

        float* Xnext = (l == NL - 1) ? out : xbuf[l & 1];
        gemm_xwT_kernel<<<gridQ, 256, 0, stream>>>(AO, outW + (size_t)l * DEMB * DEMB,
                                                   outB + (size_t)l * DEMB, Xnext, NFACE);
        Xcur = Xnext;
    }
}